// CrossLayer_6682969113347
// MI455X (gfx1250) — compile-verified
//
#include <hip/hip_runtime.h>

typedef __attribute__((ext_vector_type(2))) float v2f;
typedef __attribute__((ext_vector_type(8))) float v8f;

#define BATCH   4096
#define IN_DIM  1024
#define FD      64
#define OD      30
#define KSTR    (FD * OD)        // 1920 floats per kernel row

#define MT       16              // batch rows per workgroup
#define KCH      256             // K-chunk staged in LDS
#define XS_STRIDE 258            // padded LDS row stride (floats) to avoid bank conflicts

// ---------------------------------------------------------------------------
// Kernel 1: W[i,j] = sum_d kernel[i, f_j, d] * kernel[j, f_i, d]   (symmetric)
// 1024x1024 entries, one thread each. ~63 MFLOP, completely trivial.
// ---------------------------------------------------------------------------
__global__ __launch_bounds__(256) void build_W(const float* __restrict__ ker,
                                               const int* __restrict__ fidx,
                                               float* __restrict__ W) {
    const int idx = blockIdx.x * 256 + threadIdx.x;   // 0 .. 1024*1024-1
    const int i = idx >> 10;
    const int j = idx & 1023;
    const int fi = fidx[i];
    const int fj = fidx[j];
    const float* __restrict__ a = ker + i * KSTR + fj * OD;
    const float* __restrict__ b = ker + j * KSTR + fi * OD;
    float s = 0.f;
#pragma unroll
    for (int d = 0; d < OD; ++d) s += a[d] * b[d];
    W[idx] = s;
}

// ---------------------------------------------------------------------------
// Kernel 2: fused  Y = X @ W  (f32 WMMA 16x16x4)  +  out = 0.5*(rowsum(X.*Y)
//           - sum_i x_i^2 W_ii)
// Grid: 256 WGs (16 batch rows each) x 256 threads (8 waves).
// Wave w owns N-columns [w*128, w*128+128) as 8 held 16x16 accumulators.
// K staged through LDS in 256-wide chunks of X.
// ---------------------------------------------------------------------------
__global__ __launch_bounds__(256) void ffm_gemm(const float* __restrict__ X,
                                                const float* __restrict__ W,
                                                float* __restrict__ out) {
    __shared__ float Xs[MT * XS_STRIDE];   // ~16.5 KB
    __shared__ float red[MT];
    __shared__ float redD[MT];

    const int batch0 = blockIdx.x * MT;
    const int tid  = threadIdx.x;
    const int wave = tid >> 5;             // 0..7
    const int lane = tid & 31;

    if (tid < MT) { red[tid] = 0.f; redD[tid] = 0.f; }

    v8f acc[8];
#pragma unroll
    for (int t = 0; t < 8; ++t) acc[t] = (v8f)(0.f);

    const int nbase = wave * 128;          // this wave's N chunk
    const int m_a   = lane & 15;           // A-fragment row (16x4 f32 layout)
    const int ksel  = (lane >> 4) * 2;     // lanes 16-31 hold K+2,K+3
    const int n_b   = lane & 15;           // B-fragment column

    for (int kb = 0; kb < IN_DIM; kb += KCH) {
        __syncthreads();
        // Cooperative stage of X[batch0..+16][kb..kb+256] into LDS (float4 rows).
        for (int v = tid; v < MT * (KCH / 4); v += 256) {
            const int row  = v / (KCH / 4);
            const int col4 = v % (KCH / 4);
            const float4 src =
                ((const float4*)(X + (size_t)(batch0 + row) * IN_DIM + kb))[col4];
            float* dst = &Xs[row * XS_STRIDE + col4 * 4];
            dst[0] = src.x; dst[1] = src.y; dst[2] = src.z; dst[3] = src.w;
        }
        __syncthreads();

#pragma unroll
        for (int t = 0; t < 8; ++t) {
            const int ncol = nbase + t * 16 + n_b;
            const float* __restrict__ wp = W + ncol;
#pragma unroll 4
            for (int k = 0; k < KCH; k += 4) {
                v2f a, b;
                const float* xs = &Xs[m_a * XS_STRIDE + k + ksel]; // ds_load_b64
                a.x = xs[0];
                a.y = xs[1];
                const float* wrow = wp + (size_t)(kb + k + ksel) * IN_DIM;
                b.x = wrow[0];        // B[k+ksel  , n]  (coalesced across lanes)
                b.y = wrow[IN_DIM];   // B[k+ksel+1, n]
                acc[t] = __builtin_amdgcn_wmma_f32_16x16x4_f32(
                    /*neg_a=*/false, a, /*neg_b=*/false, b,
                    /*c_mod=*/(short)0, acc[t],
                    /*reuse_a=*/false, /*reuse_b=*/false);
            }
        }
    }

    // Fused epilogue: partial[m] += Y[m,n] * X[m,n] over this wave's columns.
    // C/D layout: VGPR r -> M = r (lanes 0-15) / r+8 (lanes 16-31); N = lane&15.
    float sums[8];
#pragma unroll
    for (int r = 0; r < 8; ++r) sums[r] = 0.f;
#pragma unroll
    for (int t = 0; t < 8; ++t) {
        const int n = nbase + t * 16 + (lane & 15);
#pragma unroll
        for (int r = 0; r < 8; ++r) {
            const int m = r + ((lane >> 4) << 3);
            sums[r] += acc[t][r] * X[(size_t)(batch0 + m) * IN_DIM + n];
        }
    }
#pragma unroll
    for (int r = 0; r < 8; ++r) {
        const int m = r + ((lane >> 4) << 3);
        atomicAdd(&red[m], sums[r]);       // ds_add_f32
    }

    // Diagonal correction: redD[m] = sum_i x[b,i]^2 * W[i,i]
    {
        const int m   = tid >> 4;          // 0..15
        const int seg = tid & 15;          // 64-wide slice of i
        const float* xr = X + (size_t)(batch0 + m) * IN_DIM + seg * 64;
        float s = 0.f;
#pragma unroll 8
        for (int i = 0; i < 64; ++i) {
            const float xv = xr[i];
            s += xv * xv * W[(size_t)(seg * 64 + i) * (IN_DIM + 1)];
        }
        atomicAdd(&redD[m], s);
    }

    __syncthreads();
    if (tid < MT) out[batch0 + tid] = 0.5f * (red[tid] - redD[tid]);
}

// ---------------------------------------------------------------------------
extern "C" void kernel_launch(void* const* d_in, const int* in_sizes, int n_in,
                              void* d_out, int out_size, void* d_ws, size_t ws_size,
                              hipStream_t stream) {
    const float* x    = (const float*)d_in[0];   // (4096, 1024) f32
    const float* ker  = (const float*)d_in[1];   // (1024, 64, 30) f32
    const int*   fidx = (const int*)d_in[2];     // (1024,) i32
    float* W   = (float*)d_ws;                   // 1024*1024*4 = 4 MB scratch
    float* outp = (float*)d_out;                 // (4096,) f32

    build_W<<<(IN_DIM * IN_DIM) / 256, 256, 0, stream>>>(ker, fidx, W);
    ffm_gemm<<<BATCH / MT, 256, 0, stream>>>(x, W, outp);
}